// TransformerBlock_10024453669018
// MI455X (gfx1250) — compile-verified
//
#include <hip/hip_runtime.h>
#include <hip/hip_bf16.h>
#include <math.h>

// ---------------------------------------------------------------------------
// Problem constants (match reference)
// ---------------------------------------------------------------------------
#define BATCH   16
#define NTOK    512          // tokens stage 1
#define DFEAT   256          // features stage 1 (tokens stage 2)
#define NHEAD   8
#define DHEAD   16
#define INNER   128          // NHEAD*DHEAD
#define QKV3    384          // 3*INNER
#define SCALE_ATTN 0.25f     // DH**-0.5

typedef float v2f __attribute__((ext_vector_type(2)));
typedef float v8f __attribute__((ext_vector_type(8)));

#define WMMA_F32(a, b, c) \
    __builtin_amdgcn_wmma_f32_16x16x4_f32(false, (a), false, (b), (short)0, (c), false, false)

// ---------------------------------------------------------------------------
// Generic WMMA fp32 GEMM:  C[M,N] = A[M,K] @ W[K,N] (+ bias[N])
// block = 128 threads (4 waves). Block tile: 64(M) x 64(N).
// Wave w owns rows w*16..w*16+15 and all 64 columns (4 accumulators); each A
// fragment is reused across 4 WMMAs.
// W panel stored pair-interleaved in LDS (K-rows 2p/2p+1 interleaved per
// column) so every B fragment is ONE aligned ds_load_b64 -> no repack movs.
// Requires: M%64==0, N%64==0, K%64==0 (true for every GEMM here).
// ---------------------------------------------------------------------------
#define GEMM_BK 64
#define SWP_STRIDE (64 * 2 + 4)                 // floats per pair-row (+pad)

__global__ __launch_bounds__(128)
void wmma_gemm(const float* __restrict__ A, const float* __restrict__ W,
               const float* __restrict__ bias, float* __restrict__ C,
               int M, int N, int K) {
    __shared__ float sA[64][GEMM_BK + 4];                  // A panel, padded
    __shared__ float sWp[GEMM_BK / 2][SWP_STRIDE];         // W panel, paired

    const int tid  = threadIdx.x;
    const int wave = tid >> 5;
    const int lane = tid & 31;
    const int hf   = lane >> 4;                 // lane half (0/1)
    const int l16  = lane & 15;

    const int row0 = blockIdx.x * 64;
    const int col0 = blockIdx.y * 64;
    const int wr   = wave * 16;                 // wave's row offset in tile

    v8f acc0 = {}, acc1 = {}, acc2 = {}, acc3 = {};

    for (int k0 = 0; k0 < K; k0 += GEMM_BK) {
        // prefetch next K-panel into cache (global_prefetch_b8)
        if (k0 + GEMM_BK < K) {
            int r = tid & 63;
            __builtin_prefetch(A + (size_t)(row0 + r) * K + k0 + GEMM_BK, 0, 3);
            __builtin_prefetch(W + (size_t)(k0 + GEMM_BK + r) * N + col0, 0, 3);
        }
        // A tile 64x64: coalesced along K
        for (int i = tid; i < 64 * GEMM_BK; i += 128) {
            int r = i >> 6, c = i & 63;
            sA[r][c] = A[(size_t)(row0 + r) * K + k0 + c];
        }
        // W tile 64x64 -> pair-interleaved: rows 2p/2p+1 interleave per column
        for (int i = tid; i < GEMM_BK * 64; i += 128) {
            int r = i >> 6, c = i & 63;
            sWp[r >> 1][c * 2 + (r & 1)] = W[(size_t)(k0 + r) * N + col0 + c];
        }
        __syncthreads();

        #pragma unroll
        for (int kk = 0; kk < GEMM_BK; kk += 4) {
            // A 16x4 fragment: one aligned b64 LDS load per lane
            const v2f a = *(const v2f*)&sA[wr + l16][kk + 2 * hf];
            // B 4x16 fragments: one aligned b64 LDS load per column group
            const int p = (kk + 2 * hf) >> 1;   // pair row index
            const v2f b0 = *(const v2f*)&sWp[p][( 0 + l16) * 2];
            const v2f b1 = *(const v2f*)&sWp[p][(16 + l16) * 2];
            const v2f b2 = *(const v2f*)&sWp[p][(32 + l16) * 2];
            const v2f b3 = *(const v2f*)&sWp[p][(48 + l16) * 2];
            acc0 = WMMA_F32(a, b0, acc0);
            acc1 = WMMA_F32(a, b1, acc1);
            acc2 = WMMA_F32(a, b2, acc2);
            acc3 = WMMA_F32(a, b3, acc3);
        }
        __syncthreads();
    }

    const int rbase = row0 + wr + hf * 8;
    #pragma unroll
    for (int cg = 0; cg < 4; ++cg) {
        const v8f acc = (cg == 0) ? acc0 : (cg == 1) ? acc1 : (cg == 2) ? acc2 : acc3;
        const int col = col0 + cg * 16 + l16;
        const float bv = bias ? bias[col] : 0.0f;
        #pragma unroll
        for (int r = 0; r < 8; ++r)
            C[(size_t)(rbase + r) * N + col] = acc[r] + bv;
    }
}

// ---------------------------------------------------------------------------
// energy[b,h,i,j] = SCALE * sum_d Q[b,i,h,d] * K[b,j,h,d]
// qkv layout: [b, tok, 384] = [q(128) | k(128) | v(128)], head h at h*16.
// block = 128 (4 waves); wave w handles j-tile blockIdx.y*64 + w*16.
// ---------------------------------------------------------------------------
__global__ __launch_bounds__(128)
void attn_energy(const float* __restrict__ qkv, float* __restrict__ energy,
                 int ntok) {
    const int bh = blockIdx.z;
    const int b  = bh >> 3;            // /NHEAD
    const int h  = bh & 7;
    const int wave = threadIdx.x >> 5;
    const int lane = threadIdx.x & 31;
    const int hf   = lane >> 4;
    const int l16  = lane & 15;

    const int i0 = blockIdx.x * 16;
    const int j0 = blockIdx.y * 64 + wave * 16;

    const float* qb = qkv + (size_t)b * ntok * QKV3 + h * DHEAD;          // Q
    const float* kb = qb + INNER;                                          // K

    v8f acc = {};
    #pragma unroll
    for (int kk = 0; kk < DHEAD; kk += 4) {
        // consecutive-address pairs -> single global b64 loads
        const v2f a     = *(const v2f*)&qb[(size_t)(i0 + l16) * QKV3 + kk + 2 * hf];
        const v2f bfrag = *(const v2f*)&kb[(size_t)(j0 + l16) * QKV3 + kk + 2 * hf];
        acc = WMMA_F32(a, bfrag, acc);
    }

    float* erow = energy + ((size_t)bh * ntok) * ntok;
    #pragma unroll
    for (int r = 0; r < 8; ++r)
        erow[(size_t)(i0 + hf * 8 + r) * ntok + j0 + l16] = acc[r] * SCALE_ATTN;
}

// ---------------------------------------------------------------------------
// Fused softmax + P@V.  One wave per 16-row tile of one (b,h).
// Probabilities live only in LDS (stride-padded) -> no 128MB probs tensor.
// ctx layout: [b, tok, INNER], head h at h*16.
// ---------------------------------------------------------------------------
__global__ __launch_bounds__(32)
void softmax_pv(const float* __restrict__ energy, const float* __restrict__ qkv,
                float* __restrict__ ctx, int ntok) {
    extern __shared__ float sp[];               // [16][ntok+4]
    const int NP = ntok + 4;

    const int bh = blockIdx.y;
    const int b  = bh >> 3;
    const int h  = bh & 7;
    const int i0 = blockIdx.x * 16;
    const int lane = threadIdx.x;
    const int hf   = lane >> 4;
    const int l16  = lane & 15;

    const float* erow = energy + ((size_t)bh * ntok + i0) * ntok;

    // exact softmax per row (max-subtracted), written into LDS
    for (int r = 0; r < 16; ++r) {
        const float* e = erow + (size_t)r * ntok;
        float mx = -INFINITY;
        for (int j = lane; j < ntok; j += 32) mx = fmaxf(mx, e[j]);
        #pragma unroll
        for (int m = 16; m; m >>= 1) mx = fmaxf(mx, __shfl_xor(mx, m, 32));
        float s = 0.0f;
        for (int j = lane; j < ntok; j += 32) {
            float p = __expf(e[j] - mx);
            sp[r * NP + j] = p;
            s += p;
        }
        #pragma unroll
        for (int m = 16; m; m >>= 1) s += __shfl_xor(s, m, 32);
        const float inv = 1.0f / s;
        for (int j = lane; j < ntok; j += 32) sp[r * NP + j] *= inv;
    }
    __syncthreads();

    // ctx[i, d] = sum_j P[i,j] * V[j,d]   (A from LDS, B from global V)
    const float* vb = qkv + (size_t)b * ntok * QKV3 + 2 * INNER + h * DHEAD;
    v8f acc = {};
    for (int kk = 0; kk < ntok; kk += 4) {
        v2f a, bfrag;
        a.x = sp[l16 * NP + kk + 2 * hf + 0];
        a.y = sp[l16 * NP + kk + 2 * hf + 1];
        bfrag.x = vb[(size_t)(kk + 2 * hf + 0) * QKV3 + l16];
        bfrag.y = vb[(size_t)(kk + 2 * hf + 1) * QKV3 + l16];
        acc = WMMA_F32(a, bfrag, acc);
    }

    #pragma unroll
    for (int r = 0; r < 8; ++r)
        ctx[(size_t)(b * ntok + i0 + hf * 8 + r) * INNER + h * DHEAD + l16] = acc[r];
}

// ---------------------------------------------------------------------------
// out[row] = LayerNorm(a[row] + res[row]) * g + b      (one block per row)
// C in {256, 512}; block = 256 threads, each owns C/256 elements.
// ---------------------------------------------------------------------------
__global__ __launch_bounds__(256)
void add_layernorm(const float* __restrict__ a, const float* __restrict__ res,
                   const float* __restrict__ g, const float* __restrict__ bt,
                   float* __restrict__ out, int C) {
    __shared__ float red[256];
    const int row = blockIdx.x;
    const int tid = threadIdx.x;
    const int cnt = C >> 8;                      // 1 or 2

    const float* pa = a   + (size_t)row * C;
    const float* pr = res + (size_t)row * C;

    float v[2];
    float s = 0.0f;
    for (int i = 0; i < cnt; ++i) {
        v[i] = pa[tid + (i << 8)] + pr[tid + (i << 8)];
        s += v[i];
    }
    red[tid] = s; __syncthreads();
    for (int st = 128; st; st >>= 1) { if (tid < st) red[tid] += red[tid + st]; __syncthreads(); }
    const float mu = red[0] / (float)C;
    __syncthreads();

    float s2 = 0.0f;
    for (int i = 0; i < cnt; ++i) { float d = v[i] - mu; s2 += d * d; }
    red[tid] = s2; __syncthreads();
    for (int st = 128; st; st >>= 1) { if (tid < st) red[tid] += red[tid + st]; __syncthreads(); }
    const float rstd = rsqrtf(red[0] / (float)C + 1e-5f);

    for (int i = 0; i < cnt; ++i) {
        int c = tid + (i << 8);
        out[(size_t)row * C + c] = (v[i] - mu) * rstd * g[c] + bt[c];
    }
}

// ---------------------------------------------------------------------------
// GEGLU: mid[r,c] = h[r,c] * gelu_exact(h[r, C + c]),  h width = 2*C
// ---------------------------------------------------------------------------
__global__ void geglu(const float* __restrict__ h, float* __restrict__ mid,
                      long total, int C) {
    long idx = (long)blockIdx.x * blockDim.x + threadIdx.x;
    long stride = (long)gridDim.x * blockDim.x;
    for (; idx < total; idx += stride) {
        long r = idx / C; int c = (int)(idx % C);
        float av = h[r * (2 * C) + c];
        float gv = h[r * (2 * C) + C + c];
        float ge = 0.5f * gv * (1.0f + erff(gv * 0.70710678118654752f));
        mid[idx] = av * ge;
    }
}

// ---------------------------------------------------------------------------
// Batched tiled transpose: in [B, R, Cc] -> out [B, Cc, R]
// block (32, 8); grid (Cc/32, R/32, B)
// ---------------------------------------------------------------------------
__global__ __launch_bounds__(256)
void transpose_b(const float* __restrict__ in, float* __restrict__ out,
                 int R, int Cc) {
    __shared__ float t[32][33];
    const int b  = blockIdx.z;
    const int c0 = blockIdx.x * 32;
    const int r0 = blockIdx.y * 32;
    const int x = threadIdx.x, y = threadIdx.y;
    const float* pin  = in  + (size_t)b * R * Cc;
    float*       pout = out + (size_t)b * R * Cc;
    #pragma unroll
    for (int i = 0; i < 32; i += 8)
        t[y + i][x] = pin[(size_t)(r0 + y + i) * Cc + c0 + x];
    __syncthreads();
    #pragma unroll
    for (int i = 0; i < 32; i += 8)
        pout[(size_t)(c0 + y + i) * R + r0 + x] = t[x][y + i];
}

// ---------------------------------------------------------------------------
// Host launcher
// ---------------------------------------------------------------------------
extern "C" void kernel_launch(void* const* d_in, const int* in_sizes, int n_in,
                              void* d_out, int out_size, void* d_ws, size_t ws_size,
                              hipStream_t stream) {
    const float* x      = (const float*)d_in[0];
    const float* wqkv1  = (const float*)d_in[1];
    const float* wo1    = (const float*)d_in[2];
    const float* bo1    = (const float*)d_in[3];
    const float* f1w1   = (const float*)d_in[4];
    const float* f1b1   = (const float*)d_in[5];
    const float* f1w2   = (const float*)d_in[6];
    const float* f1b2   = (const float*)d_in[7];
    const float* wqkv2  = (const float*)d_in[8];
    const float* wo2    = (const float*)d_in[9];
    const float* bo2    = (const float*)d_in[10];
    const float* f2w1   = (const float*)d_in[11];
    const float* f2b1   = (const float*)d_in[12];
    const float* f2w2   = (const float*)d_in[13];
    const float* f2b2   = (const float*)d_in[14];
    const float* ln1g   = (const float*)d_in[15];
    const float* ln1b   = (const float*)d_in[16];
    const float* ln2g   = (const float*)d_in[17];
    const float* ln2b   = (const float*)d_in[18];
    const float* ln3g   = (const float*)d_in[19];
    const float* ln3b   = (const float*)d_in[20];
    const float* ln4g   = (const float*)d_in[21];
    const float* ln4b   = (const float*)d_in[22];

    float* out_x   = (float*)d_out;                                // [B,N,D]
    float* share1  = out_x  + (size_t)BATCH * NTOK * DFEAT;        // [B,H,N,N]
    float* share2  = share1 + (size_t)BATCH * NHEAD * NTOK * NTOK; // [B,H,D,D]

    // workspace carve-up (floats)
    float* ws   = (float*)d_ws;
    float* qkv  = ws;                                  // 3,145,728
    float* hbuf = qkv  + (size_t)3145728;              // 16,777,216
    float* gbuf = hbuf + (size_t)16777216;             // 8,388,608
    float* ctx  = gbuf + (size_t)8388608;              // 1,048,576
    float* xa   = ctx  + (size_t)1048576;              // 2,097,152
    float* xb   = xa   + (size_t)2097152;              // 2,097,152
    float* xc   = xb   + (size_t)2097152;              // 2,097,152
    (void)ws_size; (void)n_in; (void)in_sizes; (void)out_size;

    const int M1 = BATCH * NTOK;    // 8192 rows, stage 1
    const int M2 = BATCH * DFEAT;   // 4096 rows, stage 2

    // ================= stage 1: token attention =================
    // qkv1 = x @ wqkv1                       [8192,256]x[256,384]
    wmma_gemm<<<dim3(M1 / 64, QKV3 / 64), 128, 0, stream>>>(x, wqkv1, nullptr, qkv, M1, QKV3, DFEAT);
    // energies -> share1 (scaled)
    attn_energy<<<dim3(NTOK / 16, NTOK / 64, BATCH * NHEAD), 128, 0, stream>>>(qkv, share1, NTOK);
    // fused softmax + P@V -> ctx [8192,128]
    {
        size_t shm = 16 * (NTOK + 4) * sizeof(float);
        softmax_pv<<<dim3(NTOK / 16, BATCH * NHEAD), 32, shm, stream>>>(share1, qkv, ctx, NTOK);
    }
    // proj = ctx @ wo1 + bo1                 [8192,128]x[128,256]
    wmma_gemm<<<dim3(M1 / 64, DFEAT / 64), 128, 0, stream>>>(ctx, wo1, bo1, xc, M1, DFEAT, INNER);
    // xa = LN(proj + x)
    add_layernorm<<<M1, 256, 0, stream>>>(xc, x, ln1g, ln1b, xa, DFEAT);
    // h = xa @ ff1_w1 + b1                   [8192,256]x[256,2048]
    wmma_gemm<<<dim3(M1 / 64, 2048 / 64), 128, 0, stream>>>(xa, f1w1, f1b1, hbuf, M1, 2048, DFEAT);
    // geglu -> gbuf [8192,1024]
    geglu<<<2048, 256, 0, stream>>>(hbuf, gbuf, (long)M1 * 1024, 1024);
    // ff = gbuf @ ff1_w2 + b2                [8192,1024]x[1024,256]
    wmma_gemm<<<dim3(M1 / 64, DFEAT / 64), 128, 0, stream>>>(gbuf, f1w2, f1b2, xc, M1, DFEAT, 1024);
    // xb = LN(ff + xa)
    add_layernorm<<<M1, 256, 0, stream>>>(xc, xa, ln2g, ln2b, xb, DFEAT);

    // transpose xb [B,512,256] -> xa [B,256,512]
    transpose_b<<<dim3(DFEAT / 32, NTOK / 32, BATCH), dim3(32, 8), 0, stream>>>(xb, xa, NTOK, DFEAT);

    // ================= stage 2: feature attention =================
    // qkv2 = xa @ wqkv2                      [4096,512]x[512,384]
    wmma_gemm<<<dim3(M2 / 64, QKV3 / 64), 128, 0, stream>>>(xa, wqkv2, nullptr, qkv, M2, QKV3, NTOK);
    // energies -> share2
    attn_energy<<<dim3(DFEAT / 16, DFEAT / 64, BATCH * NHEAD), 128, 0, stream>>>(qkv, share2, DFEAT);
    {
        size_t shm = 16 * (DFEAT + 4) * sizeof(float);
        softmax_pv<<<dim3(DFEAT / 16, BATCH * NHEAD), 32, shm, stream>>>(share2, qkv, ctx, DFEAT);
    }
    // proj = ctx @ wo2 + bo2                 [4096,128]x[128,512]
    wmma_gemm<<<dim3(M2 / 64, NTOK / 64), 128, 0, stream>>>(ctx, wo2, bo2, xc, M2, NTOK, INNER);
    // xb = LN(proj + xa)
    add_layernorm<<<M2, 256, 0, stream>>>(xc, xa, ln3g, ln3b, xb, NTOK);
    // h = xb @ ff2_w1 + b1                   [4096,512]x[512,4096]
    wmma_gemm<<<dim3(M2 / 64, 4096 / 64), 128, 0, stream>>>(xb, f2w1, f2b1, hbuf, M2, 4096, NTOK);
    // geglu -> gbuf [4096,2048]
    geglu<<<2048, 256, 0, stream>>>(hbuf, gbuf, (long)M2 * 2048, 2048);
    // ff = gbuf @ ff2_w2 + b2                [4096,2048]x[2048,512]
    wmma_gemm<<<dim3(M2 / 64, NTOK / 64), 128, 0, stream>>>(gbuf, f2w2, f2b2, xc, M2, NTOK, 2048);
    // xa = LN(ff + xb)
    add_layernorm<<<M2, 256, 0, stream>>>(xc, xb, ln4g, ln4b, xa, NTOK);

    // transpose back: xa [B,256,512] -> out_x [B,512,256]
    transpose_b<<<dim3(NTOK / 32, DFEAT / 32, BATCH), dim3(32, 8), 0, stream>>>(xa, out_x, DFEAT, NTOK);
}